// KWinnersBoost_78185584656737
// MI455X (gfx1250) — compile-verified
//
#include <hip/hip_runtime.h>

#define E_DIM   8192
#define THREADS 256
#define CHUNKS  8   // 8 x float4 per thread = 32 elements

// ---------------------------------------------------------------------------
// CDNA5 helpers: async global->LDS staging (gfx1250), split wait counters
// ---------------------------------------------------------------------------
__device__ __forceinline__ void async_ld_b128(unsigned lds_off, const void* gaddr) {
  asm volatile("global_load_async_to_lds_b128 %0, %1, off"
               :: "v"(lds_off), "v"((unsigned long long)(size_t)gaddr)
               : "memory");
}
__device__ __forceinline__ void wait_async0() {
#if __has_builtin(__builtin_amdgcn_s_wait_asynccnt)
  __builtin_amdgcn_s_wait_asynccnt(0);
#else
  asm volatile("s_wait_asynccnt 0" ::: "memory");
#endif
}

// monotonic float -> uint key (a > b  <=>  key(a) > key(b))
__device__ __forceinline__ unsigned f2key(float f) {
  unsigned u = __float_as_uint(f);
  return u ^ ((u & 0x80000000u) ? 0xFFFFFFFFu : 0x80000000u);
}

// ---------------------------------------------------------------------------
// ws layout (unsigned words): [0]=max bits (nonneg float), [1]=survivor count,
// [2 .. 2+2*B) = per-row {threshold key T, tie count r}
// ---------------------------------------------------------------------------
__global__ void init_ws(unsigned* ws) { ws[0] = 0u; ws[1] = 0u; }

__global__ void __launch_bounds__(256) gmax_kernel(const float4* __restrict__ x,
                                                   long n4, unsigned* ws) {
  float m = 0.0f;  // init 0 == relu clamp of the global max
  long i = (long)blockIdx.x * blockDim.x + threadIdx.x;
  long stride = (long)gridDim.x * blockDim.x;
  for (; i < n4; i += stride) {
    float4 v = x[i];
    m = fmaxf(m, fmaxf(fmaxf(v.x, v.y), fmaxf(v.z, v.w)));
  }
  __shared__ float red[256];
  red[threadIdx.x] = m;
  __syncthreads();
  for (int o = 128; o > 0; o >>= 1) {
    if ((int)threadIdx.x < o) red[threadIdx.x] = fmaxf(red[threadIdx.x], red[threadIdx.x + o]);
    __syncthreads();
  }
  if (threadIdx.x == 0) atomicMax((int*)ws, __float_as_int(red[0]));  // nonneg: int cmp == float cmp
}

// ---------------------------------------------------------------------------
// Main: one block per row. Async-stage row into LDS, radix-select k-th largest
// boosted key, write out + boost_out, accumulate survivor count.
// ---------------------------------------------------------------------------
__global__ void __launch_bounds__(THREADS) kwinners_main(
    const float* __restrict__ xg, const float* __restrict__ btg,
    const float* __restrict__ bpg, const unsigned* __restrict__ wsmax,
    unsigned* __restrict__ wscnt, unsigned* __restrict__ rowTR,
    float* __restrict__ outg, float* __restrict__ bog, int K)
{
  __shared__ float    lds_x[E_DIM];
  __shared__ float    lds_b[E_DIM];
  __shared__ unsigned hist[256];
  __shared__ unsigned scan[256];
  __shared__ unsigned bc[3];

  const int t   = threadIdx.x;
  const int row = blockIdx.x;
  const size_t base = (size_t)row * E_DIM;

  // --- async stage: thread t moves 16B chunks {k*256+t}; it later reads only
  // its own chunks, so s_wait_asynccnt alone (wave-local) is sufficient.
  {
    const char* xb = (const char*)(xg + base);
    const char* bb = (const char*)(btg + base);
    unsigned lx = (unsigned)(size_t)&lds_x[0];
    unsigned lb = (unsigned)(size_t)&lds_b[0];
#pragma unroll
    for (int k = 0; k < CHUNKS; ++k) {
      unsigned off = (unsigned)((k * 256 + t) * 16);
      async_ld_b128(lx + off, xb + off);
      async_ld_b128(lb + off, bb + off);
    }
  }

  const float mv = __uint_as_float(wsmax[0]);  // >= 0 by construction
  const float mb = mv * bpg[0];

  wait_async0();

  // --- keys (registers) + positivity bitmask
  unsigned keys[32];
  unsigned pos = 0u;
  const float4* lx4 = (const float4*)lds_x;
  const float4* lb4 = (const float4*)lds_b;
#pragma unroll
  for (int j = 0; j < CHUNKS; ++j) {
    float4 x4 = lx4[j * 256 + t];
    float4 b4 = lb4[j * 256 + t];
    float xv[4] = {x4.x, x4.y, x4.z, x4.w};
    float bv[4] = {b4.x, b4.y, b4.z, b4.w};
#pragma unroll
    for (int e = 0; e < 4; ++e) {
      int l = j * 4 + e;
      float r0 = xv[e] > 0.0f ? xv[e] : 0.0f;
      keys[l] = f2key(r0 + (bv[e] + mb));
      if (xv[e] > 0.0f) pos |= 1u << l;
    }
  }

  // --- radix select K-th largest key (4 byte passes, MSB first)
  const unsigned lane    = (unsigned)t & 31u;
  const unsigned laneLow = (1u << lane) - 1u;
  unsigned prefix = 0u;
  unsigned kRem   = (unsigned)K;
  unsigned eqCnt  = 0u;

  for (int pass = 3; pass >= 0; --pass) {
    hist[t] = 0u;
    __syncthreads();
    const int shift = pass * 8;
    const unsigned pm = (pass == 3) ? 0u : (0xFFFFFFFFu << (shift + 8));
#pragma unroll
    for (int l = 0; l < 32; ++l) {
      bool act = ((keys[l] & pm) == (prefix & pm));
      unsigned am = (unsigned)__ballot((int)act);
      if (am) {  // wave-uniform: skip empty slots fast in narrowed passes
        unsigned byte = (keys[l] >> shift) & 255u;
        unsigned m = am;
#pragma unroll
        for (int b = 0; b < 8; ++b) {
          unsigned bb2 = (unsigned)__ballot((int)((byte >> b) & 1u));
          m &= ((byte >> b) & 1u) ? bb2 : ~bb2;
        }
        // leader of each equal-byte group does one merged LDS atomic
        if (act && (m & laneLow) == 0u)
          atomicAdd(&hist[byte], (unsigned)__popc(m));
      }
    }
    __syncthreads();

    // suffix-inclusive scan over 256 bins (Hillis-Steele, reversed index)
    int rb = 255 - t;
    unsigned v = hist[t];
    scan[rb] = v;
    __syncthreads();
    for (int off = 1; off < 256; off <<= 1) {
      unsigned add = (rb >= off) ? scan[rb - off] : 0u;
      __syncthreads();
      v += add;
      scan[rb] = v;
      __syncthreads();
    }
    unsigned h = hist[t];
    unsigned sExcl = v - h;
    if (sExcl < kRem && v >= kRem) {  // unique winning bucket
      bc[0] = (unsigned)t;
      bc[1] = kRem - sExcl;
      bc[2] = h;
    }
    __syncthreads();
    prefix |= bc[0] << shift;
    kRem = bc[1];
    eqCnt = bc[2];
    __syncthreads();
  }

  const unsigned T = prefix;
  const unsigned r = kRem;

  // --- selection mask (exact top-k, index-ordered tie break)
  unsigned sel = 0u;
  if (r == eqCnt) {  // all ties selected (the overwhelmingly common case)
#pragma unroll
    for (int l = 0; l < 32; ++l)
      if (keys[l] >= T) sel |= 1u << l;
  } else {
    unsigned tm = 0u;
#pragma unroll
    for (int l = 0; l < 32; ++l) {
      if (keys[l] > T) sel |= 1u << l;
      if (keys[l] == T) tm |= 1u << l;
    }
    hist[t] = tm;
    scan[t] = 0u;
    __syncthreads();
    if (t == 0) {  // rare path: global index order is (chunk j, thread t, elem e)
      unsigned need = r;
      for (int j = 0; j < CHUNKS && need; ++j)
        for (int tt = 0; tt < 256 && need; ++tt) {
          unsigned nib = (hist[tt] >> (4 * j)) & 0xFu;
          for (int e = 0; e < 4 && need; ++e)
            if ((nib >> e) & 1u) { scan[tt] |= 1u << (4 * j + e); --need; }
        }
    }
    __syncthreads();
    sel |= scan[t];
  }

  // --- survivor count (global fallback condition)
  __syncthreads();
  scan[t] = (unsigned)__popc(sel & pos);
  __syncthreads();
  for (int o = 128; o > 0; o >>= 1) {
    if (t < o) scan[t] += scan[t + o];
    __syncthreads();
  }
  if (t == 0) {
    atomicAdd(wscnt, scan[0]);
    rowTR[2 * row]     = T;
    rowTR[2 * row + 1] = r;
  }

  // --- write out / boost_out (coalesced float4 stores; b re-read from LDS)
  float4* op  = (float4*)(outg + base);
  float4* bp4 = (float4*)(bog + base);
#pragma unroll
  for (int j = 0; j < CHUNKS; ++j) {
    float4 b4 = lb4[j * 256 + t];
    float bv[4] = {b4.x, b4.y, b4.z, b4.w};
    float ov[4], qv[4];
#pragma unroll
    for (int e = 0; e < 4; ++e) {
      int l = j * 4 + e;
      bool s = (sel >> l) & 1u;
      bool p = (pos >> l) & 1u;
      ov[e] = (s && p) ? 1.0f : 0.0f;
      qv[e] = s ? 0.0f : (bv[e] + mb);
    }
    float4 o; o.x = ov[0]; o.y = ov[1]; o.z = ov[2]; o.w = ov[3];
    float4 q; q.x = qv[0]; q.y = qv[1]; q.z = qv[2]; q.w = qv[3];
    op[j * 256 + t]  = o;
    bp4[j * 256 + t] = q;
  }
}

// ---------------------------------------------------------------------------
// Fallback: if no selected element was positive anywhere, out = mask.
// Uniform early-exit in the (always-taken) normal case.
// ---------------------------------------------------------------------------
__global__ void __launch_bounds__(THREADS) kwinners_fixup(
    const float* __restrict__ xg, const float* __restrict__ btg,
    const float* __restrict__ bpg, const unsigned* __restrict__ wsmax,
    const unsigned* __restrict__ wscnt, const unsigned* __restrict__ rowTR,
    float* __restrict__ outg)
{
  if (wscnt[0] != 0u) return;

  __shared__ unsigned hist[256];
  __shared__ unsigned scan[256];
  const int t   = threadIdx.x;
  const int row = blockIdx.x;
  const size_t base = (size_t)row * E_DIM;
  const float mv = __uint_as_float(wsmax[0]);
  const float mb = mv * bpg[0];
  const unsigned T = rowTR[2 * row];
  const unsigned r = rowTR[2 * row + 1];

  const float4* x4p = (const float4*)(xg + base);
  const float4* b4p = (const float4*)(btg + base);
  unsigned sel = 0u, tm = 0u;
#pragma unroll
  for (int j = 0; j < CHUNKS; ++j) {
    float4 x4 = x4p[j * 256 + t];
    float4 b4 = b4p[j * 256 + t];
    float xv[4] = {x4.x, x4.y, x4.z, x4.w};
    float bv[4] = {b4.x, b4.y, b4.z, b4.w};
#pragma unroll
    for (int e = 0; e < 4; ++e) {
      int l = j * 4 + e;
      float r0 = xv[e] > 0.0f ? xv[e] : 0.0f;
      unsigned k = f2key(r0 + (bv[e] + mb));
      if (k > T)  sel |= 1u << l;
      if (k == T) tm  |= 1u << l;
    }
  }
  hist[t] = (unsigned)__popc(tm);
  __syncthreads();
  for (int o = 128; o > 0; o >>= 1) {
    if (t < o) hist[t] += hist[t + o];
    __syncthreads();
  }
  unsigned eq = hist[0];
  __syncthreads();
  if (r == eq) {
    sel |= tm;
  } else {
    scan[t] = tm;
    hist[t] = 0u;
    __syncthreads();
    if (t == 0) {
      unsigned need = r;
      for (int j = 0; j < CHUNKS && need; ++j)
        for (int tt = 0; tt < 256 && need; ++tt) {
          unsigned nib = (scan[tt] >> (4 * j)) & 0xFu;
          for (int e = 0; e < 4 && need; ++e)
            if ((nib >> e) & 1u) { hist[tt] |= 1u << (4 * j + e); --need; }
        }
    }
    __syncthreads();
    sel |= hist[t];
  }

  float4* op = (float4*)(outg + base);
#pragma unroll
  for (int j = 0; j < CHUNKS; ++j) {
    float ov[4];
#pragma unroll
    for (int e = 0; e < 4; ++e) {
      int l = j * 4 + e;
      ov[e] = ((sel >> l) & 1u) ? 1.0f : 0.0f;
    }
    float4 o; o.x = ov[0]; o.y = ov[1]; o.z = ov[2]; o.w = ov[3];
    op[j * 256 + t] = o;
  }
}

// ---------------------------------------------------------------------------
extern "C" void kernel_launch(void* const* d_in, const int* in_sizes, int n_in,
                              void* d_out, int out_size, void* d_ws, size_t ws_size,
                              hipStream_t stream) {
  const float* x  = (const float*)d_in[0];
  const float* bt = (const float*)d_in[1];
  const float* bp = (const float*)d_in[2];
  const long n = (long)in_sizes[0];
  const int  E = E_DIM;
  const int  B = (int)(n / E);
  const int  K = (E + 49) / 50;  // ceil(0.02 * E) = 164 for E=8192

  float* out = (float*)d_out;
  float* bo  = out + (size_t)B * E;
  unsigned* ws    = (unsigned*)d_ws;
  unsigned* rowTR = ws + 2;

  init_ws<<<1, 1, 0, stream>>>(ws);
  gmax_kernel<<<4096, 256, 0, stream>>>((const float4*)x, n / 4, ws);
  kwinners_main<<<B, THREADS, 0, stream>>>(x, bt, bp, ws, ws + 1, rowTR, out, bo, K);
  kwinners_fixup<<<B, THREADS, 0, stream>>>(x, bt, bp, ws, ws + 1, rowTR, out);
}